// CorotationalBeam2DNormalized_42734924595228
// MI455X (gfx1250) — compile-verified
//
#include <hip/hip_runtime.h>
#include <hip/hip_bf16.h>

// Problem constants (match reference)
#define N_NODES 500000
#define N_EDGES 4000000
#define EPS_F   1e-10f

static constexpr int NF         = 3 * N_NODES;   // 1,500,000 floats per (N,3) field
static constexpr int RANGE_BASE = 3 * NF;        // float index of k_ax_range[0] in d_out

static constexpr int GRID   = 2048;
static constexpr int BLOCK  = 256;
static constexpr int STRIDE = GRID * BLOCK;      // compile-time grid stride (524288)

// ---------------------------------------------------------------------------
// Kernel 1: initialize output accumulator region + min/max slots.
// ---------------------------------------------------------------------------
__global__ void __launch_bounds__(BLOCK)
cb2d_init_kernel(float* __restrict__ out) {
    // Zero the 1.5M-float force accumulator as float4 (NF % 4 == 0).
    float4* __restrict__ out4 = reinterpret_cast<float4*>(out);
    const float4 z = make_float4(0.f, 0.f, 0.f, 0.f);
    for (int i = blockIdx.x * BLOCK + threadIdx.x; i < NF / 4; i += STRIDE) {
        out4[i] = z;
    }
    if (blockIdx.x == 0 && threadIdx.x == 0) {
        unsigned int* r = reinterpret_cast<unsigned int*>(out + RANGE_BASE);
        r[0] = 0x7f7fffffu;  // k_ax   min slot = FLT_MAX
        r[1] = 0x00000000u;  // k_ax   max slot = 0.0f   (k >= 0 always)
        r[2] = 0x7f7fffffu;  // k_bend min slot = FLT_MAX
        r[3] = 0x00000000u;  // k_bend max slot = 0.0f
    }
}

// ---------------------------------------------------------------------------
// Kernel 2: per-edge corotational beam forces, atomic scatter-add,
// block-reduced min/max of k_ax / k_bend.
// ---------------------------------------------------------------------------
__global__ void __launch_bounds__(BLOCK)
cb2d_edge_kernel(const float* __restrict__ pred,    // (N_NODES,3)
                 const float* __restrict__ coords,  // (N_NODES,3)
                 const float* __restrict__ propE,   // (N_EDGES)
                 const float* __restrict__ propA,   // (N_EDGES)
                 const float* __restrict__ propI,   // (N_EDGES)
                 const int2*  __restrict__ conn,    // (N_EDGES,2) int32 pairs
                 float*       __restrict__ forces,  // d_out[0..NF)
                 unsigned int* __restrict__ ranges) // d_out + RANGE_BASE (4 slots)
{
    __shared__ float s_red[4][BLOCK / 32];  // [quantity][wave]

    float kax_min = 3.402823466e38f, kax_max = 0.0f;
    float kbd_min = 3.402823466e38f, kbd_max = 0.0f;

    for (int e = blockIdx.x * BLOCK + threadIdx.x; e < N_EDGES; e += STRIDE) {
        // Prefetch next iteration's streaming data (global_prefetch_b8).
        __builtin_prefetch(conn  + (e + STRIDE), 0, 0);
        __builtin_prefetch(propE + (e + STRIDE), 0, 0);

        const int2 ab = conn[e];
        const int na = ab.x;
        const int nb = ab.y;

        // Gathers: node tables (~12 MB total) are L2-resident.
        const float ax = coords[3 * na + 0];
        const float az = coords[3 * na + 2];
        const float bx = coords[3 * nb + 0];
        const float bz = coords[3 * nb + 2];

        const float dx0 = bx - ax;
        const float dz0 = bz - az;
        const float l0  = sqrtf(dx0 * dx0 + dz0 * dz0 + EPS_F);
        const float inv_l = 1.0f / (l0 + EPS_F);
        const float c = dx0 * inv_l;
        const float s = dz0 * inv_l;

        const float E  = propE[e];
        const float EA = E * propA[e];
        const float EI = E * propI[e];

        const float k_ax   = EA * inv_l;
        const float k_bend = EI * inv_l;
        const float k_sw   = EI / (l0 * l0 + EPS_F);
        const float k_tr   = EI / (l0 * l0 * l0 + EPS_F);

        kax_min = fminf(kax_min, k_ax);   kax_max = fmaxf(kax_max, k_ax);
        kbd_min = fminf(kbd_min, k_bend); kbd_max = fmaxf(kbd_max, k_bend);

        const float uxA =  pred[3 * na + 0];
        const float uzA =  pred[3 * na + 1];
        const float ta  = -pred[3 * na + 2];
        const float uxB =  pred[3 * nb + 0];
        const float uzB =  pred[3 * nb + 1];
        const float tb  = -pred[3 * nb + 2];

        const float ua =  c * uxA + s * uzA;
        const float wa = -s * uxA + c * uzA;
        const float ub =  c * uxB + s * uzB;
        const float wb = -s * uxB + c * uzB;

        // f3 = -f0 and f4 = -f1  =>  fgB.x = -fgA.x, fgB.y = -fgA.y
        const float f0 = k_ax * (ua - ub);
        const float f1 = 12.0f * k_tr * (wa - wb) + 6.0f * k_sw * (ta + tb);
        const float sw_wab = 6.0f * k_sw * (wa - wb);
        const float f2 = sw_wab + k_bend * (4.0f * ta + 2.0f * tb);
        const float f5 = sw_wab + k_bend * (2.0f * ta + 4.0f * tb);

        const float gx = c * f0 - s * f1;
        const float gz = s * f0 + c * f1;

        // Scatter-add (global_atomic_add_f32, L2 atomic units).
        atomicAdd(&forces[3 * na + 0],  gx);
        atomicAdd(&forces[3 * na + 1],  gz);
        atomicAdd(&forces[3 * na + 2],  f2);
        atomicAdd(&forces[3 * nb + 0], -gx);
        atomicAdd(&forces[3 * nb + 1], -gz);
        atomicAdd(&forces[3 * nb + 2],  f5);
    }

    // ---- wave32 butterfly reduction ----
    #pragma unroll
    for (int off = 16; off > 0; off >>= 1) {
        kax_min = fminf(kax_min, __shfl_xor(kax_min, off, 32));
        kax_max = fmaxf(kax_max, __shfl_xor(kax_max, off, 32));
        kbd_min = fminf(kbd_min, __shfl_xor(kbd_min, off, 32));
        kbd_max = fmaxf(kbd_max, __shfl_xor(kbd_max, off, 32));
    }
    const int lane = threadIdx.x & 31;
    const int wave = threadIdx.x >> 5;
    if (lane == 0) {
        s_red[0][wave] = kax_min;
        s_red[1][wave] = kax_max;
        s_red[2][wave] = kbd_min;
        s_red[3][wave] = kbd_max;
    }
    __syncthreads();
    if (threadIdx.x == 0) {
        float m0 = s_red[0][0], m1 = s_red[1][0], m2 = s_red[2][0], m3 = s_red[3][0];
        #pragma unroll
        for (int w = 1; w < BLOCK / 32; ++w) {
            m0 = fminf(m0, s_red[0][w]);
            m1 = fmaxf(m1, s_red[1][w]);
            m2 = fminf(m2, s_red[2][w]);
            m3 = fmaxf(m3, s_red[3][w]);
        }
        // k values are >= 0, so uint ordering == float ordering.
        atomicMin(&ranges[0], __float_as_uint(m0));
        atomicMax(&ranges[1], __float_as_uint(m1));
        atomicMin(&ranges[2], __float_as_uint(m2));
        atomicMax(&ranges[3], __float_as_uint(m3));
    }
}

// ---------------------------------------------------------------------------
// Kernel 3: per-node maps.
//   out_fext[j] = F_ext[j]                          (float4 streaming copy)
//   out_phys[i] = pred[i] * (comp==2 ? theta_scale : u_scale)
// ---------------------------------------------------------------------------
__global__ void __launch_bounds__(BLOCK)
cb2d_node_kernel(const float* __restrict__ pred,
                 const float* __restrict__ fext,
                 const float* __restrict__ u_scale,
                 const float* __restrict__ theta_scale,
                 float* __restrict__ out_fext,
                 float* __restrict__ out_phys)
{
    const int tid = blockIdx.x * BLOCK + threadIdx.x;
    const float us = u_scale[0];
    const float ts = theta_scale[0];

    // phys_disp: coalesced elementwise; comp = i % 3 (magic-number mul, no div unit).
    for (int i = tid; i < NF; i += STRIDE) {
        const int comp = i % 3;
        out_phys[i] = pred[i] * (comp == 2 ? ts : us);
    }

    // F_ext passthrough as float4 (NF % 4 == 0; both bases 16B-aligned).
    const float4* __restrict__ src = reinterpret_cast<const float4*>(fext);
    float4* __restrict__ dst = reinterpret_cast<float4*>(out_fext);
    for (int i = tid; i < NF / 4; i += STRIDE) {
        dst[i] = src[i];
    }
}

// ---------------------------------------------------------------------------
// Launcher
// ---------------------------------------------------------------------------
extern "C" void kernel_launch(void* const* d_in, const int* in_sizes, int n_in,
                              void* d_out, int out_size, void* d_ws, size_t ws_size,
                              hipStream_t stream) {
    const float* pred   = (const float*)d_in[0];   // (N_NODES,3)
    const float* coords = (const float*)d_in[1];   // (N_NODES,3)
    const float* propE  = (const float*)d_in[2];   // (N_EDGES,)
    const float* propA  = (const float*)d_in[3];   // (N_EDGES,)
    const float* propI  = (const float*)d_in[4];   // (N_EDGES,)
    const int2*  conn   = (const int2*)d_in[5];    // (N_EDGES,2) int32
    const float* fext   = (const float*)d_in[6];   // (N_NODES,3)
    const float* uscale = (const float*)d_in[7];   // (1,)
    const float* tscale = (const float*)d_in[8];   // (1,)

    float* out        = (float*)d_out;
    float* out_forces = out;                 // [0, NF)
    float* out_fext   = out + NF;            // [NF, 2NF)
    float* out_phys   = out + 2 * NF;        // [2NF, 3NF)
    unsigned int* ranges = reinterpret_cast<unsigned int*>(out + RANGE_BASE);

    // 1) zero force accumulator + seed min/max slots
    cb2d_init_kernel<<<GRID, BLOCK, 0, stream>>>(out_forces);

    // 2) edge assembly (depends on init via stream order)
    cb2d_edge_kernel<<<GRID, BLOCK, 0, stream>>>(
        pred, coords, propE, propA, propI, conn, out_forces, ranges);

    // 3) independent per-node maps
    cb2d_node_kernel<<<GRID, BLOCK, 0, stream>>>(
        pred, fext, uscale, tscale, out_fext, out_phys);
}